// SelfModifyingLayer_74259984548275
// MI455X (gfx1250) — compile-verified
//
#include <hip/hip_runtime.h>
#include <hip/hip_bf16.h>

// CDNA5 fp32 WMMA: D(16x16 f32) = A(16x4 f32) * B(4x16 f32) + C
typedef __attribute__((ext_vector_type(2))) float v2f;
typedef __attribute__((ext_vector_type(8))) float v8f;

static __device__ __forceinline__ v8f wmma_f32(v2f a, v2f b, v8f c) {
    // 8 args: (neg_a, A, neg_b, B, c_mod, C, reuse_a, reuse_b)
    return __builtin_amdgcn_wmma_f32_16x16x4_f32(false, a, false, b, (short)0, c,
                                                 false, false);
}

// A fragment: A[m][k] row-major, lane L holds m = row0 + L%16, k = kk*4 + (L/16)*2 + v
static __device__ __forceinline__ v2f loadA(const float* base, int row0, int kk,
                                            int ld, int lane) {
    int m = row0 + (lane & 15);
    int k = kk * 4 + ((lane >> 4) << 1);
    v2f a;
    a.x = base[m * ld + k];
    a.y = base[m * ld + k + 1];
    return a;
}
// B fragment from row-major B[k][n] (leading dim ld)
static __device__ __forceinline__ v2f loadB(const float* base, int col0, int kk,
                                            int ld, int lane) {
    int n = col0 + (lane & 15);
    int k = kk * 4 + ((lane >> 4) << 1);
    v2f b;
    b.x = base[k * ld + n];
    b.y = base[(k + 1) * ld + n];
    return b;
}
// B fragment = M^T where M[n][k] is row-major (leading dim ld): B[k][n] = M[n][k]
static __device__ __forceinline__ v2f loadBT(const float* base, int col0, int kk,
                                             int ld, int lane) {
    int n = col0 + (lane & 15);
    int k = kk * 4 + ((lane >> 4) << 1);
    v2f b;
    b.x = base[n * ld + k];
    b.y = base[n * ld + k + 1];
    return b;
}

// ---------------------------------------------------------------------------
// Projection GEMM: Y[m][n] = act( sum_d X[m][d]*W[n][d] + bias[n] )
// grid = (M/64, 256/64), block = 256 (8 waves); act: 0 = identity, 1 = elu()+1
// Each wave computes a 16x32 output strip (one A row-block, two B col-blocks)
// so the A fragment is loaded once per k-step and reused for two WMMAs.
// ---------------------------------------------------------------------------
__global__ void proj_kernel(const float* __restrict__ X, const float* __restrict__ W,
                            const float* __restrict__ bias, float* __restrict__ Y,
                            int act) {
    extern __shared__ float sm[];
    float* xs = sm;              // [64][256]
    float* ws = sm + 64 * 256;   // [64][256]  (rows = output cols n, W row-major)
    const int m0 = blockIdx.x * 64;
    const int n0 = blockIdx.y * 64;
    const int tid = threadIdx.x;
    const int lane = tid & 31, wv = tid >> 5;

#pragma unroll
    for (int it = 0; it < 16; ++it) {
        int li = it * 256 + tid;               // 4096 float4 per array
        int row = li >> 6, c4 = (li & 63) << 2;
        *(float4*)&xs[row * 256 + c4] = *(const float4*)&X[(size_t)(m0 + row) * 256 + c4];
        *(float4*)&ws[row * 256 + c4] = *(const float4*)&W[(size_t)(n0 + row) * 256 + c4];
    }
    __syncthreads();

    const int ti  = wv >> 1;            // tiles (ti, ni0) and (ti, ni0+1)
    const int ni0 = (wv & 1) * 2;
    v8f c0 = {}, c1 = {};
    for (int kk = 0; kk < 64; ++kk) {
        v2f a  = loadA(xs, ti * 16, kk, 256, lane);
        v2f b0 = loadBT(ws, ni0 * 16, kk, 256, lane);        // B[d][n] = W[n][d]
        v2f b1 = loadBT(ws, (ni0 + 1) * 16, kk, 256, lane);
        c0 = wmma_f32(a, b0, c0);
        c1 = wmma_f32(a, b1, c1);
    }
    const int half = lane >> 4;
#pragma unroll
    for (int p = 0; p < 2; ++p) {
        v8f c = p ? c1 : c0;
        int n = n0 + (ni0 + p) * 16 + (lane & 15);
        float bn = bias[n];
#pragma unroll
        for (int vv = 0; vv < 8; ++vv) {
            int m = m0 + ti * 16 + vv + 8 * half;
            float y = c[vv] + bn;
            if (act) y = (y > 0.0f) ? (y + 1.0f) : __expf(y);  // elu(y)+1
            Y[(size_t)m * 256 + n] = y;
        }
    }
}

// ---------------------------------------------------------------------------
// Chunked linear-attention scan. grid = B * (256/64) = 8 blocks, block = 256.
// Each block owns state columns [e0, e0+64) of one batch and runs all 64
// chunks (C=64) sequentially:
//   out   = diag(decay^t) * (Q @ mem0)  +  (L o Q K^T) @ V
//   mem'  = decay^64 * mem0 + (diag(decay^{63-s}) K)^T @ V
// ---------------------------------------------------------------------------
__global__ void scan_kernel(const float* __restrict__ q, const float* __restrict__ k,
                            const float* __restrict__ v, const float* __restrict__ state,
                            const float* __restrict__ decay_param,
                            float* __restrict__ attn, float* __restrict__ memout) {
    extern __shared__ float sm[];
    float* memS = sm;             // [256][64]  state slice
    float* Qs   = sm + 16384;     // [64][256]
    float* Ks   = sm + 32768;     // [64][256]
    float* Vs   = sm + 49152;     // [64][64]
    float* Ss   = sm + 53248;     // [64][64]   masked scores
    float* dpow = sm + 57344;     // [64]       decay^n table

    const int b  = blockIdx.x >> 2;
    const int e0 = (blockIdx.x & 3) * 64;
    const int tid = threadIdx.x;
    const int lane = tid & 31, wv = tid >> 5;
    const int half = lane >> 4;

    const float decay  = 1.0f / (1.0f + __expf(-decay_param[0]));
    const float l2d    = __log2f(decay);
    const float decayC = exp2f(64.0f * l2d);
    if (tid < 64) dpow[tid] = exp2f((float)tid * l2d);

#pragma unroll
    for (int it = 0; it < 16; ++it) {          // init state slice (4096 float4)
        int li = it * 256 + tid;
        int d = li >> 4, c4 = (li & 15) << 2;
        *(float4*)&memS[d * 64 + c4] =
            *(const float4*)&state[(size_t)b * 65536 + d * 256 + e0 + c4];
    }
    __syncthreads();

    for (int c0i = 0; c0i < 64; ++c0i) {
        const size_t base = ((size_t)b * 4096 + (size_t)c0i * 64) * 256;

        // ---- stage Q, K (64x256) and V slice (64x64) into LDS ----
#pragma unroll
        for (int it = 0; it < 16; ++it) {
            int li = it * 256 + tid;
            int row = li >> 6, c4 = (li & 63) << 2;
            *(float4*)&Qs[row * 256 + c4] = *(const float4*)&q[base + row * 256 + c4];
            *(float4*)&Ks[row * 256 + c4] = *(const float4*)&k[base + row * 256 + c4];
        }
#pragma unroll
        for (int it = 0; it < 4; ++it) {
            int li = it * 256 + tid;
            int row = li >> 4, c4 = (li & 15) << 2;
            *(float4*)&Vs[row * 64 + c4] = *(const float4*)&v[base + row * 256 + e0 + c4];
        }
        // prefetch next chunk while this one computes (global_prefetch_b8)
        if (c0i < 63) {
            const size_t basen = base + (size_t)64 * 256;
            __builtin_prefetch(&q[basen + (size_t)tid * 64], 0, 1);
            __builtin_prefetch(&k[basen + (size_t)tid * 64], 0, 1);
            __builtin_prefetch(&v[basen + (size_t)tid * 64], 0, 1);
        }
        __syncthreads();

        // ---- S' = L o (Q K^T), 64x64, K-dim 256; 16 tiles = 8 waves x 2 ----
        {
            const int ti  = wv >> 1;
            const int si0 = (wv & 1) * 2;
            v8f s0 = {}, s1 = {};
            for (int kk = 0; kk < 64; ++kk) {
                v2f a  = loadA(Qs, ti * 16, kk, 256, lane);
                v2f b0 = loadBT(Ks, si0 * 16, kk, 256, lane);       // B[d][s]=K[s][d]
                v2f b1 = loadBT(Ks, (si0 + 1) * 16, kk, 256, lane);
                s0 = wmma_f32(a, b0, s0);
                s1 = wmma_f32(a, b1, s1);
            }
#pragma unroll
            for (int p = 0; p < 2; ++p) {
                v8f c = p ? s1 : s0;
                int s = (si0 + p) * 16 + (lane & 15);
#pragma unroll
                for (int vv = 0; vv < 8; ++vv) {
                    int t = ti * 16 + vv + 8 * half;
                    float f = (s < t) ? dpow[t - 1 - s] : 0.0f;
                    Ss[t * 64 + s] = c[vv] * f;
                }
            }
        }
        __syncthreads();

        // ---- out = decay^t * (Q @ mem0) + S' @ V; 16 tiles = 8 waves x 2 ----
        {
            const int ti  = wv >> 1;
            const int ei0 = (wv & 1) * 2;
            v8f o0 = {}, o1 = {};
            for (int kk = 0; kk < 64; ++kk) {            // inter-chunk: K-dim 256
                v2f a  = loadA(Qs, ti * 16, kk, 256, lane);
                v2f b0 = loadB(memS, ei0 * 16, kk, 64, lane);
                v2f b1 = loadB(memS, (ei0 + 1) * 16, kk, 64, lane);
                o0 = wmma_f32(a, b0, o0);
                o1 = wmma_f32(a, b1, o1);
            }
#pragma unroll
            for (int vv = 0; vv < 8; ++vv) {
                float f = dpow[ti * 16 + vv + 8 * half];
                o0[vv] *= f;
                o1[vv] *= f;
            }
            for (int kk = 0; kk < 16; ++kk) {            // intra-chunk: K-dim 64
                v2f a  = loadA(Ss, ti * 16, kk, 64, lane);
                v2f b0 = loadB(Vs, ei0 * 16, kk, 64, lane);
                v2f b1 = loadB(Vs, (ei0 + 1) * 16, kk, 64, lane);
                o0 = wmma_f32(a, b0, o0);
                o1 = wmma_f32(a, b1, o1);
            }
#pragma unroll
            for (int p = 0; p < 2; ++p) {
                v8f c = p ? o1 : o0;
                int n = (ei0 + p) * 16 + (lane & 15);
#pragma unroll
                for (int vv = 0; vv < 8; ++vv) {
                    int t = ti * 16 + vv + 8 * half;
                    attn[base + (size_t)t * 256 + e0 + n] = c[vv];
                }
            }
        }
        __syncthreads();   // all reads of old memS complete

        // ---- mem = decay^64 * mem + (diag(decay^{63-s}) K)^T @ V ----
        // 64 tiles = 8 waves x (2 A-row blocks x 4 B-col blocks); fragments
        // hoisted: per k-step 2 A + 4 B loads feed 8 WMMAs.
        {
            v8f acc[2][4];
#pragma unroll
            for (int p = 0; p < 2; ++p) {
                int di = wv * 2 + p;
#pragma unroll
                for (int e = 0; e < 4; ++e) {
                    int n = e * 16 + (lane & 15);
#pragma unroll
                    for (int vv = 0; vv < 8; ++vv) {
                        int dd = di * 16 + vv + 8 * half;
                        acc[p][e][vv] = memS[dd * 64 + n] * decayC;
                    }
                }
            }
            for (int kk = 0; kk < 16; ++kk) {            // K-dim 64 (s)
                int s0 = kk * 4 + half * 2;
                float w0 = dpow[63 - s0], w1 = dpow[62 - s0];
                v2f a[2];
#pragma unroll
                for (int p = 0; p < 2; ++p) {            // A[d][s]=K[s][d]*decay^{63-s}
                    int m = (wv * 2 + p) * 16 + (lane & 15);
                    a[p].x = Ks[s0 * 256 + m] * w0;
                    a[p].y = Ks[(s0 + 1) * 256 + m] * w1;
                }
                v2f bb[4];
#pragma unroll
                for (int e = 0; e < 4; ++e) bb[e] = loadB(Vs, e * 16, kk, 64, lane);
#pragma unroll
                for (int p = 0; p < 2; ++p)
#pragma unroll
                    for (int e = 0; e < 4; ++e)
                        acc[p][e] = wmma_f32(a[p], bb[e], acc[p][e]);
            }
#pragma unroll
            for (int p = 0; p < 2; ++p) {
                int di = wv * 2 + p;
#pragma unroll
                for (int e = 0; e < 4; ++e) {
                    int n = e * 16 + (lane & 15);
#pragma unroll
                    for (int vv = 0; vv < 8; ++vv) {
                        int dd = di * 16 + vv + 8 * half;
                        memS[dd * 64 + n] = acc[p][e][vv];
                    }
                }
            }
        }
        __syncthreads();
    }

    // ---- write final state slice ----
#pragma unroll
    for (int it = 0; it < 16; ++it) {
        int li = it * 256 + tid;
        int d = li >> 4, c4 = (li & 15) << 2;
        *(float4*)&memout[(size_t)b * 65536 + d * 256 + e0 + c4] =
            *(float4*)&memS[d * 64 + c4];
    }
}

// ---------------------------------------------------------------------------
extern "C" void kernel_launch(void* const* d_in, const int* in_sizes, int n_in,
                              void* d_out, int out_size, void* d_ws, size_t ws_size,
                              hipStream_t stream) {
    const float* x      = (const float*)d_in[0];
    const float* state  = (const float*)d_in[1];
    const float* Wq     = (const float*)d_in[2];
    const float* bq     = (const float*)d_in[3];
    const float* Wk     = (const float*)d_in[4];
    const float* bk     = (const float*)d_in[5];
    const float* Wv     = (const float*)d_in[6];
    const float* bv     = (const float*)d_in[7];
    const float* Wo     = (const float*)d_in[8];
    const float* bo     = (const float*)d_in[9];
    const float* decayp = (const float*)d_in[10];

    const size_t BSD = (size_t)2 * 4096 * 256;   // 2,097,152
    float* out      = (float*)d_out;             // [B,S,D]
    float* memfinal = out + BSD;                 // [B,D,D]

    float* ws   = (float*)d_ws;
    float* qb   = ws;
    float* kb   = ws + BSD;
    float* vb   = ws + 2 * BSD;
    float* attn = ws + 3 * BSD;

    dim3 gProj(128, 4);       // (B*S)/64 x D/64
    dim3 blk(256);
    size_t ldsProj = (size_t)2 * 64 * 256 * sizeof(float);   // 128 KB
    size_t ldsScan = (size_t)(57344 + 64) * sizeof(float);   // ~229 KB

    proj_kernel<<<gProj, blk, ldsProj, stream>>>(x, Wq, bq, qb, 0);
    proj_kernel<<<gProj, blk, ldsProj, stream>>>(x, Wk, bk, kb, 1);
    proj_kernel<<<gProj, blk, ldsProj, stream>>>(x, Wv, bv, vb, 0);

    scan_kernel<<<dim3(8), blk, ldsScan, stream>>>(qb, kb, vb, state, decayp,
                                                   attn, memfinal);

    proj_kernel<<<gProj, blk, ldsProj, stream>>>(attn, Wo, bo, out, 0);
}